// SinkhornFromScores_8830452761359
// MI455X (gfx1250) — compile-verified
//
#include <hip/hip_runtime.h>
#include <cstdint>
#include <cstddef>

// Sinkhorn (tau=1, 20 iters) as classical matvec Sinkhorn on E0 = exp(log_alpha):
//   v_i = 1 / sum_j E0_ij * u_j      (row pass)
//   u_j = 1 / sum_i E0_ij * v_i      (col pass)
//   P_ij = E0_ij * v_i * u_j
// E0 (128 MiB, L2-resident on the 192MB L2) lives in d_out and is finished in
// place. Row pass + column-partial pass are FUSED (single E0 sweep / iter).
// The 64-way column-partial reduction is done with V_WMMA_F32_16X16X4_F32
// (reduction-as-matmul with a ones B operand).

#define B    8
#define N    2048
#define N4   (N / 4)            // 512 float4 per row
#define ITER 20
#define CB   64                 // row-blocks per batch (32 rows each)

typedef __attribute__((ext_vector_type(2))) float v2f;
typedef __attribute__((ext_vector_type(8))) float v8f;

// ---------- CDNA5 async global->LDS helpers ----------
__device__ __forceinline__ unsigned lds_byte_off(const void* p) {
    // generic -> addrspace(3) cast yields the 32-bit LDS offset
    return (unsigned)(unsigned long long)(__attribute__((address_space(3))) const void*)p;
}

__device__ __forceinline__ void async_load_b128(unsigned lds_off, const void* gptr) {
    unsigned long long ga = (unsigned long long)(uintptr_t)gptr;
    asm volatile("global_load_async_to_lds_b128 %0, %1, off"
                 :: "v"(lds_off), "v"(ga)
                 : "memory");
}

__device__ __forceinline__ void wait_async0() {
    asm volatile("s_wait_asynccnt 0" ::: "memory");
}

// ---------- K_exp: E0 = exp(log_alpha) into d_out ----------
__global__ void sink_exp_kernel(const float* __restrict__ in, float* __restrict__ e0) {
    size_t q = (size_t)blockIdx.x * blockDim.x + threadIdx.x;   // float4 index
    const float4* in4 = (const float4*)in;
    float4* e4 = (float4*)e0;
    float4 a = in4[q];
    float4 r;
    r.x = __expf(a.x); r.y = __expf(a.y); r.z = __expf(a.z); r.w = __expf(a.w);
    e4[q] = r;
}

// ---------- K_init_u: u = 1 ----------
__global__ void sink_initu_kernel(float* __restrict__ u) {
    int i = blockIdx.x * blockDim.x + threadIdx.x;              // 0 .. B*N-1
    u[i] = 1.0f;
}

// ---------- Fused row-dot + column-partial kernel ----------
// grid = B*CB blocks of 256 threads; block handles 32 rows of batch b.
__global__ void sink_fused_kernel(const float* __restrict__ e0,
                                  const float* __restrict__ u,
                                  float* __restrict__ v,
                                  float* __restrict__ part) {
    __shared__ float4 u_s4[N4];           // 8 KB staged copy of u[b]
    __shared__ float  wred[2][8];         // double-buffered per-wave row partials

    const int t    = threadIdx.x;
    const int bid  = blockIdx.x;
    const int b    = bid >> 6;
    const int blk  = bid & (CB - 1);
    const int r0   = blk * 32;
    const int wave = t >> 5;
    const int lane = t & 31;

    // Stage u[b] (2048 f32) into LDS with CDNA5 async loads: 2 x b128 / thread
    {
        const float* ub = u + b * N + 8 * t;
        unsigned l0 = lds_byte_off(&u_s4[2 * t]);
        async_load_b128(l0,      ub);
        async_load_b128(l0 + 16, ub + 4);
        wait_async0();
    }
    __syncthreads();

    // Hoist this thread's u values into registers (LDS read once)
    const float4 u0 = u_s4[t];
    const float4 u1 = u_s4[t + 256];

    float4 a0 = make_float4(0.f, 0.f, 0.f, 0.f);
    float4 a1 = make_float4(0.f, 0.f, 0.f, 0.f);

    const float4* bp = (const float4*)(e0 + ((size_t)(b * N + r0)) * N);

    for (int i = 0; i < 32; ++i) {
        const float4* rp = bp + (size_t)i * N4;
        if (i < 30) __builtin_prefetch(rp + 2 * N4 + t, 0, 1);  // global_prefetch_b8

        const float4 e0v = rp[t];
        const float4 e1v = rp[t + 256];

        // row-dot partial over this thread's 8 columns
        float p = e0v.x * u0.x + e0v.y * u0.y + e0v.z * u0.z + e0v.w * u0.w
                + e1v.x * u1.x + e1v.y * u1.y + e1v.z * u1.z + e1v.w * u1.w;
#pragma unroll
        for (int m = 16; m; m >>= 1) p += __shfl_xor(p, m, 32);
        if (lane == 0) wred[i & 1][wave] = p;
        __syncthreads();

        // block-wide row sum, identical fixed order on every thread
        const float* wr = wred[i & 1];
        float S = ((wr[0] + wr[1]) + (wr[2] + wr[3]))
                + ((wr[4] + wr[5]) + (wr[6] + wr[7]));
        const float vi = 1.0f / S;
        if (t == 0) v[b * N + r0 + i] = vi;

        // column partial accumulation reusing the registers
        a0.x += e0v.x * vi; a0.y += e0v.y * vi; a0.z += e0v.z * vi; a0.w += e0v.w * vi;
        a1.x += e1v.x * vi; a1.y += e1v.y * vi; a1.z += e1v.z * vi; a1.w += e1v.w * vi;
    }

    float4* pb = (float4*)part + (size_t)(b * CB + blk) * N4;
    pb[t]       = a0;
    pb[t + 256] = a1;
}

// ---------- K_colred via WMMA: u_j = 1 / sum_blk part[b][blk][j] ----------
// One wave handles 16 columns; 16 accumulating V_WMMA_F32_16X16X4_F32 ops
// with B = ones sum all 64 partial rows: D[m][n] = sum_k A[m][k].
// A layout (ISA 7.12.2): lane L -> M = L%16, K = 2*(L/16) + {v0, v1}.
// grid = 128 blocks of 256 threads (8 waves -> 1024 tiles of 16 columns).
__global__ void sink_colred_wmma_kernel(const float* __restrict__ part,
                                        float* __restrict__ u) {
    const int t    = threadIdx.x;
    const int wave = t >> 5;
    const int lane = t & 31;
    const int g    = blockIdx.x * 8 + wave;      // tile id, 0..1023
    const int b    = g >> 7;                     // 128 tiles per batch
    const int j0   = (g & 127) * 16;
    const int M    = lane & 15;
    const int Kb   = (lane >> 4) * 2;

    v2f ones; ones[0] = 1.0f; ones[1] = 1.0f;
    v8f c = {};

    const float* pbase = part + (size_t)b * CB * N + j0 + M;
#pragma unroll
    for (int kk = 0; kk < 16; ++kk) {
        const float* pk = pbase + (size_t)(4 * kk + Kb) * N;
        v2f a; a[0] = pk[0]; a[1] = pk[N];       // K = Kb, Kb+1
        c = __builtin_amdgcn_wmma_f32_16x16x4_f32(
                /*neg_a=*/false, a, /*neg_b=*/false, ones,
                /*c_mod=*/(short)0, c, /*reuse_a=*/false, /*reuse_b=*/false);
    }

    // D layout: rows 0-7 in c[0..7] on lanes 0-15; rows 8-15 on lanes 16-31.
    // All D columns identical (B = ones), so lane 0 / lane 16 suffice.
    float* ub = u + b * N + j0;
    if (lane == 0) {
        ub[0] = 1.0f / c[0]; ub[1] = 1.0f / c[1];
        ub[2] = 1.0f / c[2]; ub[3] = 1.0f / c[3];
        ub[4] = 1.0f / c[4]; ub[5] = 1.0f / c[5];
        ub[6] = 1.0f / c[6]; ub[7] = 1.0f / c[7];
    } else if (lane == 16) {
        ub[8]  = 1.0f / c[0]; ub[9]  = 1.0f / c[1];
        ub[10] = 1.0f / c[2]; ub[11] = 1.0f / c[3];
        ub[12] = 1.0f / c[4]; ub[13] = 1.0f / c[5];
        ub[14] = 1.0f / c[6]; ub[15] = 1.0f / c[7];
    }
}

// ---------- K_final: in-place P = E0 * v_i * u_j ----------
__global__ void sink_final_kernel(float* __restrict__ e0_out,
                                  const float* __restrict__ u,
                                  const float* __restrict__ v) {
    size_t q = (size_t)blockIdx.x * blockDim.x + threadIdx.x;   // float4 index
    const int colg = (int)(q & (N4 - 1));
    const int row  = (int)((q >> 9) & (N - 1));
    const int b    = (int)(q >> 20);                            // 2^20 float4s per batch

    float4* o4 = (float4*)e0_out;
    const float4* u4 = (const float4*)u;

    float4 e  = o4[q];
    float  vi = v[b * N + row];
    float4 uu = u4[b * N4 + colg];
    float4 r;
    r.x = e.x * vi * uu.x;
    r.y = e.y * vi * uu.y;
    r.z = e.z * vi * uu.z;
    r.w = e.w * vi * uu.w;
    o4[q] = r;
}

extern "C" void kernel_launch(void* const* d_in, const int* in_sizes, int n_in,
                              void* d_out, int out_size, void* d_ws, size_t ws_size,
                              hipStream_t stream) {
    (void)in_sizes; (void)n_in; (void)out_size; (void)ws_size;

    const float* log_alpha = (const float*)d_in[0];
    float* e0 = (float*)d_out;                    // E0 lives in d_out, finished in place

    // workspace layout (floats): v[B*N] | u[B*N] | part[B*CB*N]  (~4.2 MB)
    float* v    = (float*)d_ws;
    float* u    = v + B * N;
    float* part = u + B * N;

    const int nf4 = (B * N * N) / 4;              // 8388608 float4 groups

    sink_exp_kernel  <<<nf4 / 256, 256, 0, stream>>>(log_alpha, e0);
    sink_initu_kernel<<<(B * N) / 256, 256, 0, stream>>>(u);

    for (int it = 0; it < ITER; ++it) {
        sink_fused_kernel      <<<B * CB, 256, 0, stream>>>(e0, u, v, part);
        sink_colred_wmma_kernel<<<128,    256, 0, stream>>>(part, u);
    }

    sink_final_kernel<<<nf4 / 256, 256, 0, stream>>>(e0, u, v);
}